// Transformer_16_16174846836767
// MI455X (gfx1250) — compile-verified
//
#include <hip/hip_runtime.h>
#include <math.h>

typedef __attribute__((ext_vector_type(2))) float v2f;
typedef __attribute__((ext_vector_type(8))) float v8f;

#define LN_EPS 1e-5f

// One 16x16 fp32 tile GEMM on a single wave32 using V_WMMA_F32_16X16X4_F32.
// A(m,k): ta ? A[k*lda+m] : A[m*lda+k]   (m = lane&15 for lanes of the A fragment)
// B(k,n): tb ? B[n*ldb+k] : B[k*ldb+n]
// Fragment layout (ISA 7.12.2, 32-bit A 16x4 / B 4x16):
//   lanes 0-15 hold K = k0+{0,1}, lanes 16-31 hold K = k0+{2,3}
__device__ __forceinline__ v8f gemm16_tile(const float* A, int lda, bool ta,
                                           const float* B, int ldb, bool tb,
                                           int K, int lane, v8f c) {
  const int mn = lane & 15;
  const int kh = (lane >> 4) << 1;
  for (int k0 = 0; k0 < K; k0 += 4) {
    const int ka = k0 + kh;
    v2f a, b;
    a.x = ta ? A[(ka    ) * lda + mn] : A[mn * lda + ka    ];
    a.y = ta ? A[(ka + 1) * lda + mn] : A[mn * lda + ka + 1];
    b.x = tb ? B[mn * ldb + ka    ] : B[(ka    ) * ldb + mn];
    b.y = tb ? B[mn * ldb + ka + 1] : B[(ka + 1) * ldb + mn];
    c = __builtin_amdgcn_wmma_f32_16x16x4_f32(false, a, false, b, (short)0, c,
                                              false, false);
  }
  return c;
}

__global__ __launch_bounds__(256)
void fused_transformer16(const float* __restrict__ t,
                         const float* __restrict__ gat_W,
                         const float* __restrict__ a_src,
                         const float* __restrict__ a_dst,
                         const float* __restrict__ te_W,
                         const float* __restrict__ te_b,
                         const float* __restrict__ pe,
                         const float* __restrict__ qkv_W,
                         const float* __restrict__ qkv_b,
                         const float* __restrict__ out_W,
                         const float* __restrict__ out_b,
                         const float* __restrict__ ln1_g,
                         const float* __restrict__ ln1_b,
                         const float* __restrict__ ff1_W,
                         const float* __restrict__ ff1_b,
                         const float* __restrict__ ff2_W,
                         const float* __restrict__ ff2_b,
                         const float* __restrict__ ln2_g,
                         const float* __restrict__ ln2_b,
                         const float* __restrict__ an_W,
                         const float* __restrict__ an_b,
                         const float* __restrict__ pr_W,
                         const float* __restrict__ pr_b,
                         float* __restrict__ out) {
  __shared__ float s_h[3 * 256];      // h:      (w, n, D)
  __shared__ float s_alpha[3 * 256];  // alpha:  (w, src, dst)
  __shared__ float s_go[3 * 256];     // gat_out / "rows": (w*16+d, D) = 48x16
  __shared__ float s_xe[48 * 16];     // activation 48x16
  __shared__ float s_res[48 * 16];    // residual+proj scratch 48x16
  __shared__ float s_qkv[48 * 48];    // qkv 48x48
  __shared__ float s_ctx[48 * 16];    // attention context 48x16
  __shared__ float s_ff[48 * 64];     // ff1 output 48x64
  __shared__ float s_esrc[48];
  __shared__ float s_edst[48];
  __shared__ float4 s_lat4[192];      // latent, 16B aligned for b128 LDS reads

  const int tid = threadIdx.x;
  const int wave = tid >> 5;
  const int lane = tid & 31;
  const int hi = lane >> 4;
  const int col = lane & 15;

  // ---- Prefetch decoder weights (480KB) into L2 while we compute everything
  // else: turns the one big HBM read into overlapped latency. ----
  for (int i = tid; i < 3072; i += 256)  // pr_W: 384KB = 3072 x 128B lines
    __builtin_prefetch(pr_W + i * 32, 0, 1);
  for (int i = tid; i < 768; i += 256)   // an_W: 96KB
    __builtin_prefetch(an_W + i * 32, 0, 1);

  // ================= GAT =================
  // h[w][n][d] = sum_i x[w][n][i] * gat_W[i][d]   (K=3, scalar - trivial)
  for (int idx = tid; idx < 768; idx += 256) {
    const int w = idx >> 8, rem = idx & 255, n = rem >> 4, d = rem & 15;
    const float* xr = t + (w * 16 + n) * 3;
    s_h[idx] = xr[0] * gat_W[d] + xr[1] * gat_W[16 + d] + xr[2] * gat_W[32 + d];
  }
  __syncthreads();

  if (tid < 48) {  // e_src / e_dst: dot(h_row, a)
    float es = 0.f, ed = 0.f;
    for (int d = 0; d < 16; ++d) {
      const float hv = s_h[tid * 16 + d];
      es += hv * a_src[d];
      ed += hv * a_dst[d];
    }
    s_esrc[tid] = es;
    s_edst[tid] = ed;
  }
  __syncthreads();

  if (tid < 48) {  // softmax over sources per (w, dst)
    const int w = tid >> 4, dcol = tid & 15;
    const float ed = s_edst[w * 16 + dcol];
    float mx = -1e30f;
    float ev[16];
    for (int s = 0; s < 16; ++s) {
      float e = s_esrc[w * 16 + s] + ed;
      e = e > 0.f ? e : 0.2f * e;  // leaky_relu(0.2)
      ev[s] = e;
      mx = e > mx ? e : mx;
    }
    float sum = 0.f;
    for (int s = 0; s < 16; ++s) {
      const float ex = __expf(ev[s] - mx);
      ev[s] = ex;
      sum += ex;
    }
    const float inv = 1.f / sum;
    for (int s = 0; s < 16; ++s) s_alpha[w * 256 + s * 16 + dcol] = ev[s] * inv;
  }
  __syncthreads();

  // gat_out[w][d][f] = sum_s alpha[w][s][d] * h[w][s][f]  -> alpha^T @ h, WMMA
  for (int j = wave; j < 3; j += 8) {
    v8f c = {};
    c = gemm16_tile(s_alpha + j * 256, 16, /*ta=*/true,
                    s_h + j * 256, 16, /*tb=*/false, 16, lane, c);
#pragma unroll
    for (int i = 0; i < 8; ++i) {
      const int r = i + 8 * hi;
      float v = c[i];
      v = v > 0.f ? v : (__expf(v) - 1.f);  // elu
      s_go[j * 256 + r * 16 + col] = v;
    }
  }
  __syncthreads();

  // xe = gat_out @ te_W^T + te_b + pe   (48x16 @ 16x16, WMMA, 3 tiles)
  for (int j = wave; j < 3; j += 8) {
    v8f c = {};
    c = gemm16_tile(s_go + j * 256, 16, false, te_W, 16, /*tb=*/true, 16, lane, c);
#pragma unroll
    for (int i = 0; i < 8; ++i) {
      const int r = i + 8 * hi;
      s_xe[j * 256 + r * 16 + col] = c[i] + te_b[col] + pe[j * 16 + col];
    }
  }
  __syncthreads();

  // ================= 2 transformer encoder layers =================
  for (int l = 0; l < 2; ++l) {
    // qkv = xe @ qkv_W^T + qkv_b : 48x48 output, 9 WMMA tiles
    for (int j = wave; j < 9; j += 8) {
      const int m = j / 3, nt = j % 3;
      v8f c = {};
      c = gemm16_tile(s_xe + m * 256, 16, false,
                      qkv_W + l * 768 + nt * 256, 16, true, 16, lane, c);
#pragma unroll
      for (int i = 0; i < 8; ++i) {
        const int r = i + 8 * hi;
        s_qkv[(m * 16 + r) * 48 + nt * 16 + col] =
            c[i] + qkv_b[l * 48 + nt * 16 + col];
      }
    }
    __syncthreads();

    // attention: per (batch, head, query) — 96 tiny jobs, scalar
    if (tid < 96) {
      const int ss = tid % 3, hh = (tid / 3) % 2, b = tid / 6;
      const float* q = s_qkv + (ss * 16 + b) * 48 + hh * 8;
      float sc[3], mx = -1e30f;
      for (int ks = 0; ks < 3; ++ks) {
        const float* k = s_qkv + (ks * 16 + b) * 48 + 16 + hh * 8;
        float d = 0.f;
        for (int e = 0; e < 8; ++e) d += q[e] * k[e];
        d *= 0.35355339059327373f;  // 1/sqrt(8)
        sc[ks] = d;
        mx = d > mx ? d : mx;
      }
      float sum = 0.f;
      for (int ks = 0; ks < 3; ++ks) {
        sc[ks] = __expf(sc[ks] - mx);
        sum += sc[ks];
      }
      const float inv = 1.f / sum;
      for (int e = 0; e < 8; ++e) {
        float acc = 0.f;
        for (int ks = 0; ks < 3; ++ks)
          acc += sc[ks] * s_qkv[(ks * 16 + b) * 48 + 32 + hh * 8 + e];
        s_ctx[(ss * 16 + b) * 16 + hh * 8 + e] = acc * inv;
      }
    }
    __syncthreads();

    // out projection + residual: 3 WMMA tiles
    for (int j = wave; j < 3; j += 8) {
      v8f c = {};
      c = gemm16_tile(s_ctx + j * 256, 16, false,
                      out_W + l * 256, 16, true, 16, lane, c);
#pragma unroll
      for (int i = 0; i < 8; ++i) {
        const int r = i + 8 * hi;
        s_res[j * 256 + r * 16 + col] =
            c[i] + out_b[l * 16 + col] + s_xe[j * 256 + r * 16 + col];
      }
    }
    __syncthreads();

    if (tid < 48) {  // LayerNorm 1
      const float* p = s_res + tid * 16;
      float m = 0.f;
      for (int d = 0; d < 16; ++d) m += p[d];
      m *= 0.0625f;
      float v = 0.f;
      for (int d = 0; d < 16; ++d) { const float q = p[d] - m; v += q * q; }
      const float rs = rsqrtf(v * 0.0625f + LN_EPS);
      for (int d = 0; d < 16; ++d)
        s_xe[tid * 16 + d] = (p[d] - m) * rs * ln1_g[l * 16 + d] + ln1_b[l * 16 + d];
    }
    __syncthreads();

    // ff1 + relu: 48x64, 12 WMMA tiles
    for (int j = wave; j < 12; j += 8) {
      const int m = j >> 2, nt = j & 3;
      v8f c = {};
      c = gemm16_tile(s_xe + m * 256, 16, false,
                      ff1_W + l * 1024 + nt * 256, 16, true, 16, lane, c);
#pragma unroll
      for (int i = 0; i < 8; ++i) {
        const int r = i + 8 * hi;
        float v = c[i] + ff1_b[l * 64 + nt * 16 + col];
        s_ff[(m * 16 + r) * 64 + nt * 16 + col] = v > 0.f ? v : 0.f;
      }
    }
    __syncthreads();

    // ff2 + residual: 48x64 @ 64x16, 3 WMMA tiles (K=64 -> 16 chained WMMAs)
    for (int j = wave; j < 3; j += 8) {
      v8f c = {};
      c = gemm16_tile(s_ff + j * 1024, 64, false,
                      ff2_W + l * 1024, 64, true, 64, lane, c);
#pragma unroll
      for (int i = 0; i < 8; ++i) {
        const int r = i + 8 * hi;
        s_res[j * 256 + r * 16 + col] =
            c[i] + ff2_b[l * 16 + col] + s_xe[j * 256 + r * 16 + col];
      }
    }
    __syncthreads();

    if (tid < 48) {  // LayerNorm 2
      const float* p = s_res + tid * 16;
      float m = 0.f;
      for (int d = 0; d < 16; ++d) m += p[d];
      m *= 0.0625f;
      float v = 0.f;
      for (int d = 0; d < 16; ++d) { const float q = p[d] - m; v += q * q; }
      const float rs = rsqrtf(v * 0.0625f + LN_EPS);
      for (int d = 0; d < 16; ++d)
        s_xe[tid * 16 + d] = (p[d] - m) * rs * ln2_g[l * 16 + d] + ln2_b[l * 16 + d];
    }
    __syncthreads();
  }

  // latent[b*48 + s*16 + f] = xe[s][b][f]  (batch-major flatten)
  {
    float* s_lat = (float*)s_lat4;
    for (int idx = tid; idx < 768; idx += 256) {
      const int b = idx / 48, rem = idx % 48, ss = rem >> 4, f = rem & 15;
      s_lat[idx] = s_xe[(ss * 16 + b) * 16 + f];
    }
  }
  __syncthreads();

  // Decoder heads: 160 dot-products of length 768 (weights hot in L2 from the
  // prefetch); float4 weight loads -> global_load_b128, latent via ds b128.
  if (tid < 160) {
    const int o = (tid < 32) ? tid : tid - 32;
    const float* W = (tid < 32) ? (an_W + o * 768) : (pr_W + o * 768);
    const float4* W4 = (const float4*)W;
    float acc = 0.f;
#pragma unroll 8
    for (int k = 0; k < 192; ++k) {
      const float4 w = W4[k];
      const float4 l = s_lat4[k];
      acc += w.x * l.x + w.y * l.y + w.z * l.z + w.w * l.w;
    }
    if (tid < 32) {
      out[tid] = acc + an_b[o];  // leaky_relu slope 1.0 == identity
    } else {
      const float z = acc + pr_b[o];
      out[tid] = 1.f / (1.f + __expf(-z));  // sigmoid
    }
  }
}

extern "C" void kernel_launch(void* const* d_in, const int* in_sizes, int n_in,
                              void* d_out, int out_size, void* d_ws, size_t ws_size,
                              hipStream_t stream) {
  (void)in_sizes; (void)n_in; (void)d_ws; (void)ws_size; (void)out_size;
  const float* t      = (const float*)d_in[0];
  // d_in[1] = 's' (unused by the reference computation)
  const float* gat_W  = (const float*)d_in[2];
  const float* a_src  = (const float*)d_in[3];
  const float* a_dst  = (const float*)d_in[4];
  const float* te_W   = (const float*)d_in[5];
  const float* te_b   = (const float*)d_in[6];
  const float* pe     = (const float*)d_in[7];
  const float* qkv_W  = (const float*)d_in[8];
  const float* qkv_b  = (const float*)d_in[9];
  const float* out_W  = (const float*)d_in[10];
  const float* out_b  = (const float*)d_in[11];
  const float* ln1_g  = (const float*)d_in[12];
  const float* ln1_b  = (const float*)d_in[13];
  const float* ff1_W  = (const float*)d_in[14];
  const float* ff1_b  = (const float*)d_in[15];
  const float* ff2_W  = (const float*)d_in[16];
  const float* ff2_b  = (const float*)d_in[17];
  const float* ln2_g  = (const float*)d_in[18];
  const float* ln2_b  = (const float*)d_in[19];
  const float* an_W   = (const float*)d_in[20];
  const float* an_b   = (const float*)d_in[21];
  const float* pr_W   = (const float*)d_in[22];
  const float* pr_b   = (const float*)d_in[23];

  fused_transformer16<<<1, 256, 0, stream>>>(
      t, gat_W, a_src, a_dst, te_W, te_b, pe, qkv_W, qkv_b, out_W, out_b,
      ln1_g, ln1_b, ff1_W, ff1_b, ff2_W, ff2_b, ln2_g, ln2_b,
      an_W, an_b, pr_W, pr_b, (float*)d_out);
}